// GNN_29274497089560
// MI455X (gfx1250) — compile-verified
//
#include <hip/hip_runtime.h>
#include <hip/hip_bf16.h>

#define IN_FEAT 128
#define HIDDEN  64

typedef __attribute__((ext_vector_type(2))) float v2f;
typedef __attribute__((ext_vector_type(8))) float v8f;

// ---------------------------------------------------------------------------
// deg[i] = 1.0 (self-loop weight), then atomic-accumulate edge weights by col
// ---------------------------------------------------------------------------
__global__ void init_deg_kernel(float* __restrict__ deg, int n) {
    int i = blockIdx.x * blockDim.x + threadIdx.x;
    if (i < n) deg[i] = 1.0f;
}

__global__ void deg_accum_kernel(const int* __restrict__ cols,
                                 const float* __restrict__ ew,
                                 float* __restrict__ deg, int e) {
    int i = blockIdx.x * blockDim.x + threadIdx.x;
    if (i < e) atomicAdd(&deg[cols[i]], ew[i]);
}

__global__ void dinv_kernel(float* __restrict__ deg, int n) {
    int i = blockIdx.x * blockDim.x + threadIdx.x;
    if (i < n) {
        float d = deg[i];
        deg[i] = (d > 0.0f) ? __frsqrt_rn(d) : 0.0f;  // in place: deg -> dinv
    }
}

// ---------------------------------------------------------------------------
// H0 = X (N x 128) @ W1 (128 x 64) via V_WMMA_F32_16X16X4_F32.
// One wave computes one 16x16 tile of H0; 8 waves / block; W1 staged in LDS.
// A 16x4 f32 layout: lanes 0..15 = M rows; VGPR0 = K{0 | 2}, VGPR1 = K{1 | 3}
//   selected by lane-half. B mirrors with lanes = N columns. C/D: 8 VGPRs,
//   lane<16 -> (M=v, N=lane), lane>=16 -> (M=v+8, N=lane-16).
// ---------------------------------------------------------------------------
__global__ void __launch_bounds__(256)
gemm1_wmma_kernel(const float* __restrict__ X, const float* __restrict__ W,
                  float* __restrict__ H, int n) {
    __shared__ float sW[IN_FEAT * HIDDEN];   // 32 KB of the 320 KB/WGP LDS
    for (int i = threadIdx.x; i < IN_FEAT * HIDDEN; i += blockDim.x)
        sW[i] = W[i];
    __syncthreads();

    const int wave = blockIdx.x * (blockDim.x >> 5) + (threadIdx.x >> 5);
    const int lane = threadIdx.x & 31;
    const int mt = wave >> 2;           // M tile (16 rows each)
    const int nt = wave & 3;            // N tile (4 tiles of 16 cols)
    const int m0 = mt * 16;
    if (m0 >= n) return;
    const int n0 = nt * 16;
    const int half = lane >> 4;         // 0: K+0/K+1, 1: K+2/K+3
    const int l = lane & 15;

    const float* Xrow = X + (size_t)(m0 + l) * IN_FEAT;
    v8f c = {0.f, 0.f, 0.f, 0.f, 0.f, 0.f, 0.f, 0.f};

#pragma unroll 8
    for (int k = 0; k < IN_FEAT; k += 4) {
        v2f a = *(const v2f*)(Xrow + k + 2 * half);           // global_load_b64
        v2f b;
        b.x = sW[(k + 2 * half + 0) * HIDDEN + n0 + l];       // ds_load_b32
        b.y = sW[(k + 2 * half + 1) * HIDDEN + n0 + l];
        c = __builtin_amdgcn_wmma_f32_16x16x4_f32(
                /*neg_a=*/false, a, /*neg_b=*/false, b,
                /*c_mod=*/(short)0, c, /*reuse_a=*/false, /*reuse_b=*/false);
    }

    float* Hout = H + (size_t)(m0 + 8 * half) * HIDDEN + n0 + l;
#pragma unroll
    for (int v = 0; v < 8; v++)
        Hout[(size_t)v * HIDDEN] = c[v];
}

// ---------------------------------------------------------------------------
// agg[i][f] = dinv[i]^2 * H0[i][f]   (self-loop term, also serves as init)
// one wave per node; each lane handles 2 features (float2)
// ---------------------------------------------------------------------------
__global__ void init_agg_kernel(const float* __restrict__ dinv,
                                const float* __restrict__ h0,
                                float* __restrict__ agg, int n) {
    int wave = blockIdx.x * (blockDim.x >> 5) + (threadIdx.x >> 5);
    int lane = threadIdx.x & 31;
    if (wave >= n) return;
    float d = dinv[wave];
    float s = d * d;
    const float2* src = (const float2*)(h0 + (size_t)wave * HIDDEN);
    float2* dst = (float2*)(agg + (size_t)wave * HIDDEN);
    float2 v = src[lane];
    v.x *= s; v.y *= s;
    dst[lane] = v;
}

// ---------------------------------------------------------------------------
// Edge scatter, layer 1: agg[col] += (dinv[row]*ew*dinv[col]) * H0[row]
// one wave per edge; lane = 2 features; f32 global atomics
// ---------------------------------------------------------------------------
__global__ void scatter1_kernel(const int* __restrict__ rows,
                                const int* __restrict__ cols,
                                const float* __restrict__ ew,
                                const float* __restrict__ dinv,
                                const float* __restrict__ h0,
                                float* __restrict__ agg, int e) {
    int wave = blockIdx.x * (blockDim.x >> 5) + (threadIdx.x >> 5);
    int lane = threadIdx.x & 31;
    if (wave >= e) return;
    int r = rows[wave];
    int c = cols[wave];
    float norm = dinv[r] * ew[wave] * dinv[c];
    float2 v = ((const float2*)(h0 + (size_t)r * HIDDEN))[lane];
    float* dst = agg + (size_t)c * HIDDEN + 2 * lane;
    atomicAdd(dst + 0, norm * v.x);
    atomicAdd(dst + 1, norm * v.y);
}

// ---------------------------------------------------------------------------
// h = relu(agg + b1), in place over N*64 elements
// ---------------------------------------------------------------------------
__global__ void relu_bias_kernel(float* __restrict__ agg,
                                 const float* __restrict__ b1, int total) {
    int i = blockIdx.x * blockDim.x + threadIdx.x;
    if (i < total) {
        float v = agg[i] + b1[i & (HIDDEN - 1)];
        agg[i] = fmaxf(v, 0.0f);
    }
}

// ---------------------------------------------------------------------------
// g[i] = dot(h[i,:], w2);  out[i] = b2 + dinv[i]^2 * g[i]   (self-loop init)
// one wave per node; lane = 2 features; wave32 shuffle reduction
// ---------------------------------------------------------------------------
__global__ void gemv2_init_kernel(const float* __restrict__ h,
                                  const float* __restrict__ w2,
                                  const float* __restrict__ b2,
                                  const float* __restrict__ dinv,
                                  float* __restrict__ g,
                                  float* __restrict__ out, int n) {
    int wave = blockIdx.x * (blockDim.x >> 5) + (threadIdx.x >> 5);
    int lane = threadIdx.x & 31;
    if (wave >= n) return;
    float2 hv = ((const float2*)(h + (size_t)wave * HIDDEN))[lane];
    float2 wv = ((const float2*)w2)[lane];
    float p = hv.x * wv.x + hv.y * wv.y;
#pragma unroll
    for (int off = 16; off > 0; off >>= 1)
        p += __shfl_xor(p, off, 32);
    if (lane == 0) {
        float d = dinv[wave];
        g[wave] = p;
        out[wave] = b2[0] + d * d * p;
    }
}

// ---------------------------------------------------------------------------
// Edge scatter, layer 2: out[col] += norm * g[row]   (scalar atomics)
// ---------------------------------------------------------------------------
__global__ void scatter2_kernel(const int* __restrict__ rows,
                                const int* __restrict__ cols,
                                const float* __restrict__ ew,
                                const float* __restrict__ dinv,
                                const float* __restrict__ g,
                                float* __restrict__ out, int e) {
    int i = blockIdx.x * blockDim.x + threadIdx.x;
    if (i >= e) return;
    int r = rows[i];
    int c = cols[i];
    float norm = dinv[r] * ew[i] * dinv[c];
    atomicAdd(&out[c], norm * g[r]);
}

// ---------------------------------------------------------------------------
extern "C" void kernel_launch(void* const* d_in, const int* in_sizes, int n_in,
                              void* d_out, int out_size, void* d_ws, size_t ws_size,
                              hipStream_t stream) {
    const float* x  = (const float*)d_in[0];
    const int*   ei = (const int*)d_in[1];
    const float* ew = (const float*)d_in[2];
    const float* w1 = (const float*)d_in[3];
    const float* b1 = (const float*)d_in[4];
    const float* w2 = (const float*)d_in[5];
    const float* b2 = (const float*)d_in[6];
    float* out = (float*)d_out;

    const int E = in_sizes[2];
    const int N = in_sizes[0] / IN_FEAT;
    const int* rows = ei;        // edge_index[0]
    const int* cols = ei + E;    // edge_index[1]

    float* ws   = (float*)d_ws;
    float* dinv = ws;                                  // N floats (deg -> dinv)
    float* g    = ws + N;                              // N floats
    float* h0   = ws + 2 * (size_t)N;                  // N*64 floats
    float* agg  = h0 + (size_t)N * HIDDEN;             // N*64 floats

    const int B = 256;
    // degree + normalization
    init_deg_kernel<<<(N + B - 1) / B, B, 0, stream>>>(dinv, N);
    deg_accum_kernel<<<(E + B - 1) / B, B, 0, stream>>>(cols, ew, dinv, E);
    dinv_kernel<<<(N + B - 1) / B, B, 0, stream>>>(dinv, N);

    // layer 1 dense GEMM (WMMA): H0 = X @ W1
    {
        int mtiles = (N + 15) / 16;
        int waves  = mtiles * (HIDDEN / 16);
        int blocks = (waves + 7) / 8;
        gemm1_wmma_kernel<<<blocks, B, 0, stream>>>(x, w1, h0, N);
    }

    // layer 1 aggregation
    init_agg_kernel<<<(N + 7) / 8, B, 0, stream>>>(dinv, h0, agg, N);
    scatter1_kernel<<<(E + 7) / 8, B, 0, stream>>>(rows, cols, ew, dinv, h0, agg, E);
    relu_bias_kernel<<<((size_t)N * HIDDEN + B - 1) / B, B, 0, stream>>>(agg, b1, N * HIDDEN);

    // layer 2: GEMV + self-loop init, then scalar edge scatter
    gemv2_init_kernel<<<(N + 7) / 8, B, 0, stream>>>(agg, w2, b2, dinv, g, out, N);
    scatter2_kernel<<<(E + B - 1) / B, B, 0, stream>>>(rows, cols, ew, dinv, g, out, E);
}